// HardTripletLoss_44933947851025
// MI455X (gfx1250) — compile-verified
//
#include <hip/hip_runtime.h>
#include <math.h>

#define BSZ    512
#define DIM    256
#define MARGIN 0.3f

typedef __attribute__((ext_vector_type(2))) float v2f;
typedef __attribute__((ext_vector_type(8))) float v8f;

// ---------------------------------------------------------------------------
// Zero the global accumulators (total loss f32, triplet count i32).
// ---------------------------------------------------------------------------
__global__ void init_accum_kernel(float* g_total, int* g_cnt) {
  if (threadIdx.x == 0 && blockIdx.x == 0) {
    *g_total = 0.0f;
    *g_cnt   = 0;
  }
}

// ---------------------------------------------------------------------------
// Squared L2 norm of each embedding row. One wave32 per row.
// ---------------------------------------------------------------------------
__global__ void norms_kernel(const float* __restrict__ E, float* __restrict__ norms) {
  int wid  = (blockIdx.x * blockDim.x + threadIdx.x) >> 5;  // global wave id = row
  int lane = threadIdx.x & 31;
  if (wid >= BSZ) return;
  const float* row = E + wid * DIM;
  float s = 0.0f;
  for (int k = lane; k < DIM; k += 32) {
    float v = row[k];
    s = fmaf(v, v, s);
  }
  for (int off = 16; off; off >>= 1) s += __shfl_xor(s, off, 32);
  if (lane == 0) norms[wid] = s;
}

// ---------------------------------------------------------------------------
// Gram matrix G = E * E^T via V_WMMA_F32_16X16X4_F32, fused distance epilogue:
//   D[m][n] = sqrt(max(0, |x_m|^2 + |x_n|^2 - 2 G[m][n])), diag forced to 0.
// One wave per 16x16 output tile; 64 WMMAs along K=256.
//
// Fragment layouts per CDNA5 ISA 7.12.2:
//   A (16x4 f32, 2 VGPRs): lanes 0-15 -> M=lane, {K=k0,k0+1};
//                          lanes 16-31 -> M=lane-16, {K=k0+2,k0+3}
//   B (4x16 f32, 2 VGPRs): lanes 0-15 -> N=lane, {K=k0,k0+1};
//                          lanes 16-31 -> N=lane-16, {K=k0+2,k0+3}
//   C/D (16x16 f32, 8 VGPRs): N = lane%16, M = r + 8*(lane>=16), r = 0..7
// ---------------------------------------------------------------------------
__global__ void gram_dist_kernel(const float* __restrict__ E,
                                 const float* __restrict__ norms,
                                 float* __restrict__ D) {
  int gwave = (blockIdx.x * blockDim.x + threadIdx.x) >> 5;  // 0..1023
  int lane  = threadIdx.x & 31;
  int tm = gwave >> 5;          // tile row 0..31
  int tn = gwave & 31;          // tile col 0..31
  int m0 = tm * 16;
  int n0 = tn * 16;

  int half = lane >> 4;         // 0: K pair {0,1}; 1: K pair {2,3}
  int l    = lane & 15;
  const float* arow = E + (m0 + l) * DIM;   // A-fragment source row
  const float* brow = E + (n0 + l) * DIM;   // B-fragment source row (E^T column)

  v8f acc = {};
  for (int k = 0; k < DIM; k += 4) {
    int ka = k + half * 2;
    v2f a = *(const v2f*)(arow + ka);   // global_load_b64
    v2f b = *(const v2f*)(brow + ka);
    acc = __builtin_amdgcn_wmma_f32_16x16x4_f32(
        /*neg_a=*/false, a, /*neg_b=*/false, b,
        /*c_mod=*/(short)0, acc, /*reuse_a=*/false, /*reuse_b=*/false);
  }

  int   n  = n0 + l;
  float nn = norms[n];
  #pragma unroll
  for (int r = 0; r < 8; ++r) {
    int   m  = m0 + r + half * 8;
    float g  = acc[r];
    float sq = norms[m] + nn - 2.0f * g;
    float d  = (sq <= 0.0f) ? 0.0f : sqrtf(sq);
    if (m == n) d = 0.0f;               // exact-zero diagonal like the reference
    D[m * BSZ + n] = d;
  }
}

// ---------------------------------------------------------------------------
// Per-anchor semi-hard scan (the O(B^3) part). One block per anchor i.
// The anchor's distance row (2KB) and the label vector live in LDS; each
// thread owns a strided set of positives j and scans all k.
// ---------------------------------------------------------------------------
__global__ void loss_kernel(const float* __restrict__ D,
                            const int*   __restrict__ labels,
                            float* __restrict__ g_total,
                            int*   __restrict__ g_cnt) {
  __shared__ float rowD[BSZ];
  __shared__ int   lab[BSZ];
  __shared__ float s_wmin[8];
  __shared__ float s_wtot[8];
  __shared__ int   s_wcnt[8];
  __shared__ float s_minneg;

  int i    = blockIdx.x;
  int tid  = threadIdx.x;
  int lane = tid & 31;
  int wid  = tid >> 5;

  const float* drow = D + (size_t)i * BSZ;
  for (int k = tid; k < BSZ; k += blockDim.x) {
    rowD[k] = drow[k];
    lab[k]  = labels[k];
  }
  __syncthreads();

  int lab_i = lab[i];

  // min over negatives (nonm includes the diagonal, matching the reference)
  float mn = INFINITY;
  for (int k = tid; k < BSZ; k += blockDim.x) {
    bool nm = (lab[k] != lab_i) || (k == i);
    mn = fminf(mn, nm ? rowD[k] : INFINITY);
  }
  for (int off = 16; off; off >>= 1) mn = fminf(mn, __shfl_xor(mn, off, 32));
  if (lane == 0) s_wmin[wid] = mn;
  __syncthreads();
  if (tid == 0) {
    float m = s_wmin[0];
    for (int w = 1; w < 8; ++w) m = fminf(m, s_wmin[w]);
    s_minneg = m;
  }
  __syncthreads();
  float min_neg = s_minneg;

  float total = 0.0f;
  int   ntrip = 0;
  for (int j = tid; j < BSZ; j += blockDim.x) {
    bool match = (lab[j] == lab_i) && (j != i);   // == valid (has_pos implied)
    if (!match) continue;
    float pdm = rowD[j];
    float hi  = pdm + MARGIN;
    int   cnt = 0;
    float sh  = 0.0f;
    for (int k = 0; k < BSZ; ++k) {
      float dn = rowD[k];
      bool  nm = (lab[k] != lab_i) || (k == i);
      if (nm && (dn < hi) && (dn > pdm)) {
        ++cnt;
        sh += (pdm - dn + MARGIN);                // relu is identity in-band
      }
    }
    float fb = fmaxf(pdm - min_neg + MARGIN, 0.0f);
    if (cnt > 0)       { total += sh; ntrip += cnt; }
    else if (fb > 0.f) { total += fb; ntrip += 1; }
  }

  for (int off = 16; off; off >>= 1) {
    total += __shfl_xor(total, off, 32);
    ntrip += __shfl_xor(ntrip, off, 32);
  }
  if (lane == 0) { s_wtot[wid] = total; s_wcnt[wid] = ntrip; }
  __syncthreads();
  if (tid == 0) {
    float t = 0.0f; int c = 0;
    for (int w = 0; w < 8; ++w) { t += s_wtot[w]; c += s_wcnt[w]; }
    atomicAdd(g_total, t);   // global_atomic_add_f32
    atomicAdd(g_cnt, c);     // global_atomic_add_u32
  }
}

// ---------------------------------------------------------------------------
// out = ntrip > 0 ? total / ntrip : 0
// ---------------------------------------------------------------------------
__global__ void finalize_kernel(const float* g_total, const int* g_cnt, float* out) {
  if (threadIdx.x == 0 && blockIdx.x == 0) {
    int c  = *g_cnt;
    out[0] = (c > 0) ? (*g_total / (float)c) : 0.0f;
  }
}

// ---------------------------------------------------------------------------
extern "C" void kernel_launch(void* const* d_in, const int* in_sizes, int n_in,
                              void* d_out, int out_size, void* d_ws, size_t ws_size,
                              hipStream_t stream) {
  const float* E      = (const float*)d_in[0];   // f32[512,256]
  const int*   labels = (const int*)d_in[1];     // i32[512]
  float*       out    = (float*)d_out;           // f32 scalar

  char*  ws      = (char*)d_ws;
  float* norms   = (float*)ws;                                  // 512 f32
  float* D       = (float*)(ws + 4096);                         // 512*512 f32 = 1MB
  float* g_total = (float*)(ws + 4096 + (size_t)BSZ * BSZ * 4); // f32
  int*   g_cnt   = (int*)  (ws + 4096 + (size_t)BSZ * BSZ * 4 + 16);

  init_accum_kernel<<<1, 32, 0, stream>>>(g_total, g_cnt);
  // 512 rows, 8 waves per 256-thread block -> 64 blocks
  norms_kernel<<<64, 256, 0, stream>>>(E, norms);
  // 32x32 = 1024 tiles, 8 waves per block -> 128 blocks
  gram_dist_kernel<<<128, 256, 0, stream>>>(E, norms, D);
  loss_kernel<<<BSZ, 256, 0, stream>>>(D, labels, g_total, g_cnt);
  finalize_kernel<<<1, 32, 0, stream>>>(g_total, g_cnt, out);
}